// RationalBasicBlock_74010876445220
// MI455X (gfx1250) — compile-verified
//
#include <hip/hip_runtime.h>

// ---------------------------------------------------------------------------
// Problem constants (from reference: B=64, C=3, H=W=512, fp32 NCHW)
// ---------------------------------------------------------------------------
constexpr int BN_    = 64;
constexpr int CH_    = 3;
constexpr int HH_    = 512;
constexpr int WW_    = 512;
constexpr int HW_    = HH_ * WW_;        // 262144
constexpr int CHW_   = CH_ * HW_;        // 786432
constexpr int TOTAL_ = BN_ * CHW_;       // 50331648
constexpr float EPS_ = 1e-5f;
constexpr float NPC_ = 16777216.0f;      // elements per channel = B*H*W

// Tiling for the WMMA conv kernels
constexpr int TH_       = 8;             // tile rows  (one wave per row)
constexpr int TW_       = 128;           // tile cols  (8 m-tiles of 16 px)
constexpr int TILES_X_  = WW_ / TW_;     // 4
constexpr int TILES_Y_  = HH_ / TH_;     // 64
constexpr int TILES_IMG_= TILES_X_ * TILES_Y_;  // 256
constexpr int NTILES_   = BN_ * TILES_IMG_;     // 16384
constexpr int GBLK_     = 2048;                 // conv grid size
constexpr int TILES_BLK_= NTILES_ / GBLK_;      // 8 tiles per block (uniform)

constexpr int LDS_RS_ = TW_ + 4;                 // 132 (padded row stride)
constexpr int LDS_CS_ = (TH_ + 2) * LDS_RS_;     // 1320 per channel
constexpr int FILL_N_ = CH_ * (TH_ + 2) * (TW_ + 2);  // 3900 elements to stage

typedef __attribute__((ext_vector_type(2))) float v2f;
typedef __attribute__((ext_vector_type(8))) float v8f;

// ---------------------------------------------------------------------------
// Rational activation  P(x)/(1+|Q(x)|), degP=5, degQ=4 (no const term)
// ---------------------------------------------------------------------------
__device__ __forceinline__ float rational_eval(float x,
                                               const float* __restrict__ a,
                                               const float* __restrict__ b) {
  float P = fmaf(x, fmaf(x, fmaf(x, fmaf(x, fmaf(x, a[5], a[4]), a[3]), a[2]), a[1]), a[0]);
  float q = x * fmaf(x, fmaf(x, fmaf(x, b[3], b[2]), b[1]), b[0]);
  return P / (1.0f + fabsf(q));
}

// ---------------------------------------------------------------------------
// Implicit-GEMM 3x3 conv (C=3 -> C=3) using V_WMMA_F32_16X16X4_F32.
//   M = 16 output pixels (one row segment), N = 16 (3 real out-channels),
//   K = 27 (c*9 + ky*3 + kx) padded to 28 -> 7 chained WMMAs.
// PREOP==1 applies y = rational(scale[c]*x + bias[c]) while staging into LDS
// (fuses BN1 + activation into conv2's input path, applied once per element).
// Also produces deterministic per-block partial sums/sumsq of the conv output
// for the following batch-norm.
// ---------------------------------------------------------------------------
template<int PREOP>
__global__ __launch_bounds__(256)
void conv_wmma(const float* __restrict__ in, const float* __restrict__ wgt,
               float* __restrict__ outp, float* __restrict__ partial,
               const float* __restrict__ stats,
               const float* __restrict__ ca0, const float* __restrict__ cb0,
               const float* __restrict__ ca1, const float* __restrict__ cb1,
               const float* __restrict__ ca2, const float* __restrict__ cb2) {
  __shared__ float lds[CH_ * LDS_CS_];   // 3960 floats (~15.8 KB)
  __shared__ float redsum[64];
  __shared__ float redsq[64];

  const int tid  = threadIdx.x;
  const int wave = tid >> 5;             // 0..7  (wave32)
  const int lane = tid & 31;
  const int idx  = lane & 15;            // A: pixel M / B,D: column N
  const int half = lane >> 4;            // K-pair selector per ISA layout

  // BN scale/bias for the pre-op (PREOP==1 only)
  float psc0 = 0.f, psc1 = 0.f, psc2 = 0.f, pbi0 = 0.f, pbi1 = 0.f, pbi2 = 0.f;
  if constexpr (PREOP) {
    psc0 = stats[0]; psc1 = stats[1]; psc2 = stats[2];
    pbi0 = stats[4]; pbi1 = stats[5]; pbi2 = stats[6];
  }

  // ---- B fragments: weights as 28x16 matrix (zero-padded), per ISA layout:
  //      element K of each group-of-4 lives at vgpr (K%2), lane-half (K/2).
  v2f bf[7];
#pragma unroll
  for (int g = 0; g < 7; ++g) {
    float be0 = 0.f, be1 = 0.f;
#pragma unroll
    for (int e = 0; e < 2; ++e) {
      int k = 4 * g + 2 * half + e;
      float v = 0.f;
      if (k < 27 && idx < CH_) {
        int c = k / 9, r = (k % 9) / 3, s = k % 3;
        v = wgt[idx * 27 + c * 9 + r * 3 + s];   // OIHW
      }
      if (e == 0) be0 = v; else be1 = v;
    }
    bf[g].x = be0; bf[g].y = be1;
  }

  // ---- Per-lane LDS offsets for the A fragments (constant across tiles)
  int aoff[7][2];
#pragma unroll
  for (int g = 0; g < 7; ++g) {
#pragma unroll
    for (int e = 0; e < 2; ++e) {
      int k = 4 * g + 2 * half + e;
      if (k > 26) k = 0;                 // k==27 pad: B row is zero anyway
      int c = k / 9, r = (k % 9) / 3, s = k % 3;
      aoff[g][e] = c * LDS_CS_ + r * LDS_RS_ + s;
    }
  }
  const int rbase = wave * LDS_RS_ + idx;

  float ssum = 0.f, ssq = 0.f;           // per-lane stats (lanes idx<3)

  for (int it = 0; it < TILES_BLK_; ++it) {
    const int tile = blockIdx.x + it * GBLK_;        // uniform per block
    const int img  = tile / TILES_IMG_;
    const int trem = tile % TILES_IMG_;
    const int ty   = trem / TILES_X_;
    const int tx   = trem % TILES_X_;
    const int h0   = ty * TH_;
    const int w0   = tx * TW_;
    const float* ibase = in + img * CHW_;

    // ---- Stage (TH+2)x(TW+2)x3 input tile (with halo, zero-padded borders)
    for (int t = tid; t < FILL_N_; t += 256) {
      int c   = t / ((TH_ + 2) * (TW_ + 2));
      int rr  = t - c * ((TH_ + 2) * (TW_ + 2));
      int row = rr / (TW_ + 2);
      int col = rr - row * (TW_ + 2);
      int gh  = h0 + row - 1;
      int gw  = w0 + col - 1;
      float v = 0.f;
      if ((unsigned)gh < (unsigned)HH_ && (unsigned)gw < (unsigned)WW_)
        v = ibase[c * HW_ + gh * WW_ + gw];
      if constexpr (PREOP) {
        float sc = (c == 0) ? psc0 : ((c == 1) ? psc1 : psc2);
        float bi = (c == 0) ? pbi0 : ((c == 1) ? pbi1 : pbi2);
        const float* ap = (c == 0) ? ca0 : ((c == 1) ? ca1 : ca2);
        const float* bp = (c == 0) ? cb0 : ((c == 1) ? cb1 : cb2);
        v = rational_eval(fmaf(v, sc, bi), ap, bp);
      }
      lds[c * LDS_CS_ + row * LDS_RS_ + col] = v;
    }
    __syncthreads();

    // ---- wave `wave` computes output row h0+wave, 8 m-tiles of 16 pixels
    float* orow = outp + img * CHW_ + (h0 + wave) * WW_ + w0;
#pragma unroll
    for (int mt = 0; mt < TW_ / 16; ++mt) {
      const int cb = rbase + mt * 16;
      v8f acc = {0.f, 0.f, 0.f, 0.f, 0.f, 0.f, 0.f, 0.f};
#pragma unroll
      for (int g = 0; g < 7; ++g) {
        v2f a;
        a.x = lds[aoff[g][0] + cb];
        float ay = lds[aoff[g][1] + cb];
        if (g == 6 && half == 1) ay = 0.f;           // K=27 padding lane
        a.y = ay;
        acc = __builtin_amdgcn_wmma_f32_16x16x4_f32(
            false, a, false, bf[g], (short)0, acc, false, false);
      }
      // D layout: lane holds channel N=idx, pixels M = v + 8*half (consecutive)
      if (idx < CH_) {
        float* op = orow + idx * HW_ + mt * 16 + half * 8;
        float4 p0; p0.x = acc[0]; p0.y = acc[1]; p0.z = acc[2]; p0.w = acc[3];
        float4 p1; p1.x = acc[4]; p1.y = acc[5]; p1.z = acc[6]; p1.w = acc[7];
        *reinterpret_cast<float4*>(op)     = p0;
        *reinterpret_cast<float4*>(op + 4) = p1;
#pragma unroll
        for (int v = 0; v < 8; ++v) { float d = acc[v]; ssum += d; ssq = fmaf(d, d, ssq); }
      }
    }
    __syncthreads();
  }

  // ---- deterministic block-level reduction of BN statistics
  if (idx < CH_) {
    redsum[wave * 8 + half * 4 + idx] = ssum;
    redsq [wave * 8 + half * 4 + idx] = ssq;
  }
  __syncthreads();
  if (tid < CH_) {
    float s = 0.f, q = 0.f;
#pragma unroll
    for (int i = 0; i < 16; ++i) { s += redsum[i * 4 + tid]; q += redsq[i * 4 + tid]; }
    partial[blockIdx.x * 8 + tid]     = s;
    partial[blockIdx.x * 8 + 4 + tid] = q;
  }
}

// ---------------------------------------------------------------------------
// Reduce per-block partials -> per-channel BN scale/bias:
//   scale = gamma * rsqrt(var+eps), bias = beta - mean*scale
// ---------------------------------------------------------------------------
__global__ __launch_bounds__(192)
void finalize_stats(const float* __restrict__ partial,
                    const float* __restrict__ gamma,
                    const float* __restrict__ beta,
                    float* __restrict__ sb) {
  __shared__ float tmp[6 * 32];
  const int tid = threadIdx.x;
  const int g = tid >> 5, l = tid & 31;
  if (g < 6) {
    int q = g / 3, ch = g % 3;
    float s = 0.f;
    for (int j = l; j < GBLK_; j += 32) s += partial[j * 8 + q * 4 + ch];
    tmp[g * 32 + l] = s;
  }
  __syncthreads();
  if (tid < 6) {
    float s = 0.f;
#pragma unroll
    for (int i = 0; i < 32; ++i) s += tmp[tid * 32 + i];
    tmp[tid * 32] = s;
  }
  __syncthreads();
  if (tid < CH_) {
    float sum  = tmp[tid * 32];
    float sq   = tmp[(3 + tid) * 32];
    float mean = sum / NPC_;
    float var  = sq / NPC_ - mean * mean;
    float sc   = gamma[tid] * rsqrtf(var + EPS_);
    sb[tid]     = sc;
    sb[4 + tid] = beta[tid] - mean * sc;
  }
}

// ---------------------------------------------------------------------------
// Final elementwise: out = rational_ch( BN2(conv2) + x ), vectorized float4.
// ---------------------------------------------------------------------------
__global__ __launch_bounds__(256)
void final_pointwise(const float* __restrict__ t3, const float* __restrict__ xin,
                     const float* __restrict__ sb, float* __restrict__ outp,
                     const float* __restrict__ ca0, const float* __restrict__ cb0,
                     const float* __restrict__ ca1, const float* __restrict__ cb1,
                     const float* __restrict__ ca2, const float* __restrict__ cb2) {
  size_t i4 = (size_t)blockIdx.x * 256 + threadIdx.x;
  size_t i  = i4 * 4;
  int ch = (int)((i / (size_t)HW_) % 3u);       // HW_ % 4 == 0: uniform per vec4
  float sc = sb[ch], bi = sb[4 + ch];
  const float* ap = (ch == 0) ? ca0 : ((ch == 1) ? ca1 : ca2);
  const float* bp = (ch == 0) ? cb0 : ((ch == 1) ? cb1 : cb2);
  float4 t  = reinterpret_cast<const float4*>(t3)[i4];
  float4 xv = reinterpret_cast<const float4*>(xin)[i4];
  float4 r;
  r.x = rational_eval(fmaf(t.x, sc, bi) + xv.x, ap, bp);
  r.y = rational_eval(fmaf(t.y, sc, bi) + xv.y, ap, bp);
  r.z = rational_eval(fmaf(t.z, sc, bi) + xv.z, ap, bp);
  r.w = rational_eval(fmaf(t.w, sc, bi) + xv.w, ap, bp);
  reinterpret_cast<float4*>(outp)[i4] = r;
}

// ---------------------------------------------------------------------------
// Launch pipeline.  Workspace layout (floats):
//   [0, 16384)        partial stats for BN1
//   [16384, 32768)    partial stats for BN2
//   [32768, 32776)    BN1 scale[3]/pad/bias[3]/pad
//   [32776, 32784)    BN2 scale/bias
//   [40960, ...)      conv2 output buffer (50,331,648 floats)
// d_out doubles as the conv1-output buffer (fully overwritten at the end).
// ---------------------------------------------------------------------------
extern "C" void kernel_launch(void* const* d_in, const int* in_sizes, int n_in,
                              void* d_out, int out_size, void* d_ws, size_t ws_size,
                              hipStream_t stream) {
  (void)in_sizes; (void)n_in; (void)out_size; (void)ws_size;
  const float* x  = (const float*)d_in[0];
  const float* w1 = (const float*)d_in[1];
  const float* g1 = (const float*)d_in[2];
  const float* b1 = (const float*)d_in[3];
  const float* ar = (const float*)d_in[4];
  const float* br = (const float*)d_in[5];
  const float* ag = (const float*)d_in[6];
  const float* bg = (const float*)d_in[7];
  const float* ab = (const float*)d_in[8];
  const float* bb = (const float*)d_in[9];
  const float* w2 = (const float*)d_in[10];
  const float* g2 = (const float*)d_in[11];
  const float* b2 = (const float*)d_in[12];
  float* out = (float*)d_out;
  float* wsf = (float*)d_ws;
  float* part1 = wsf;
  float* part2 = wsf + GBLK_ * 8;
  float* sb1   = wsf + 2 * GBLK_ * 8;     // 8 floats
  float* sb2   = sb1 + 8;                 // 8 floats
  float* t3    = wsf + 40960;             // conv2 output

  // conv1(x) -> out(t1), stats1
  conv_wmma<0><<<GBLK_, 256, 0, stream>>>(x, w1, out, part1, nullptr,
                                          nullptr, nullptr, nullptr, nullptr,
                                          nullptr, nullptr);
  finalize_stats<<<1, 192, 0, stream>>>(part1, g1, b1, sb1);
  // conv2( rational(BN1(t1)) ) -> t3, stats2
  conv_wmma<1><<<GBLK_, 256, 0, stream>>>(out, w2, t3, part2, sb1,
                                          ar, br, ag, bg, ab, bb);
  finalize_stats<<<1, 192, 0, stream>>>(part2, g2, b2, sb2);
  // out = rational( BN2(t3) + x )
  final_pointwise<<<TOTAL_ / 1024, 256, 0, stream>>>(t3, x, sb2, out,
                                                     ar, br, ag, bg, ab, bb);
}